// IsoTransform_72954314489945
// MI455X (gfx1250) — compile-verified
//
#include <hip/hip_runtime.h>
#include <math.h>

// Problem constants (match reference)
#define NROW 2048
#define DIN  512
#define DOUT 256
#define TOPK_K 20
#define INV_ITERS  32   // Newton-Schulz inverse (even -> result back in start buffer)
#define SQRT_ITERS 20   // Newton-Schulz sqrtm (even)

typedef float v2f __attribute__((ext_vector_type(2)));
typedef float v8f __attribute__((ext_vector_type(8)));
typedef unsigned int u32x4 __attribute__((ext_vector_type(4)));
typedef unsigned int u32x8 __attribute__((ext_vector_type(8)));

// ---------------------------------------------------------------------------
// Tensor Data Mover: 2D tile load Global -> LDS (CDNA5 ISA ch.8 descriptor).
//  - group0: count=1 | lds_addr | global_addr[56:0] | type=2
//  - group1: data_size=4B, pad_enable (pad_amount DWORDs after every
//            2^(pad_interval+1) DWORDs -> padded LDS row stride), tile dims,
//            tensor_dim0_stride = row stride in elements.
//  Issued by one wave; completion tracked with TENSORcnt (s_wait_tensorcnt),
//  published to the workgroup with a barrier.
//  LDS offset = low 32 bits of the generic pointer (LDS aperture rule).
// ---------------------------------------------------------------------------
__device__ __forceinline__ void tdm_load_2d(uint32_t lds_base, const float* gsrc,
                                            uint32_t tile_w, uint32_t tile_h,
                                            uint64_t stride_elems,
                                            uint32_t pad_interval_code,
                                            uint32_t pad_amount_code)
{
    const uint64_t ga = (uint64_t)(uintptr_t)gsrc;
    u32x4 g0;
    g0[0] = 1u;                                           // count=1 (user mode)
    g0[1] = lds_base;                                     // lds_addr (bytes)
    g0[2] = (uint32_t)ga;                                 // global_addr[31:0]
    g0[3] = (uint32_t)((ga >> 32) & 0x01FFFFFFu) | (2u << 30); // addr[56:32] | type=2

    const uint32_t dim0 = 0x40000000u;  // huge tensor dims: tiles never clip
    const uint32_t dim1 = 0x40000000u;
    u32x8 g1;
    g1[0] = (2u << 16)                      // data_size = 4 bytes
          | (1u << 20)                      // pad_enable
          | (pad_interval_code << 22)
          | (pad_amount_code << 25);
    g1[1] = (dim0 & 0xFFFFu) << 16;         // atomic_barrier_addr=0 | dim0[15:0]
    g1[2] = (dim0 >> 16) | ((dim1 & 0xFFFFu) << 16);
    g1[3] = (dim1 >> 16) | (tile_w << 16);  // tile_dim0 = tile width (elements)
    g1[4] = tile_h;                         // tile_dim1 = rows; tile_dim2 = 0
    g1[5] = (uint32_t)stride_elems;         // tensor_dim0_stride[31:0]
    g1[6] = (uint32_t)(stride_elems >> 32) & 0xFFFFu; // stride[47:32]; dim1_stride=0
    g1[7] = 0u;
    asm volatile("tensor_load_to_lds %0, %1" :: "s"(g0), "s"(g1) : "memory");
}

// ---------------------------------------------------------------------------
// Tiled fp32 WMMA GEMM:  C = alpha * A(MxK) * op(B) + beta * C + gamma * I
//   op(B) = B (KxN row major)        when TRANS_B == false
//   op(B) = B^T with B stored (NxK)  when TRANS_B == true
// 128 threads (4 waves). Block tile 64x64; each wave owns a 32x32 quadrant =
// four independent 16x16 v_wmma_f32_16x16x4_f32 accumulators (2x fragment
// reuse, 4 independent WMMA chains). K staged in 32-chunks via TDM into
// double-buffered LDS: the next chunk's DMA is issued before computing the
// current one, overlapping tensor-DMA with WMMA via TENSORcnt.
// All dims here are multiples of 64/32 -> no bounds checks.
// ---------------------------------------------------------------------------
template<bool TRANS_B>
__global__ __launch_bounds__(128, 2)
void gemm_wmma_f32(const float* __restrict__ A, int lda,
                   const float* __restrict__ B, int ldb,
                   float* __restrict__ C, int ldc,
                   int K, float alpha, float beta, float gamma)
{
    // [m][k], stride 36 (TDM pads 32+4 DW per row); double buffered
    __shared__ float As[2][64][36];
    // NT: [n][k] stride 36; NN: [k][n] stride 68 (TDM pads 64+4); double buffered
    __shared__ float Bsmem[2][TRANS_B ? (64 * 36) : (32 * 68)];

    const int tid  = threadIdx.x;
    const int wave = tid >> 5;          // wave32
    const int lane = tid & 31;
    const int hi   = lane >> 4;         // lane half 0/1
    const int lm   = lane & 15;
    const int wm   = (wave >> 1) << 5;  // wave quadrant in 64x64 tile
    const int wn   = (wave & 1)  << 5;
    const int tm   = blockIdx.y << 6;
    const int tn   = blockIdx.x << 6;

    v8f acc00 = {0.f,0.f,0.f,0.f,0.f,0.f,0.f,0.f};
    v8f acc01 = acc00, acc10 = acc00, acc11 = acc00;

    // issue both TDM tile loads for chunk kb into LDS buffer `buf`
    auto issue_chunk = [&](int buf, int kb) {
        tdm_load_2d((uint32_t)(uintptr_t)&As[buf][0][0],
                    A + (size_t)tm * lda + kb, 32u, 64u, (uint64_t)lda, 4u, 3u);
        if (TRANS_B) {
            tdm_load_2d((uint32_t)(uintptr_t)&Bsmem[buf][0],
                        B + (size_t)tn * ldb + kb, 32u, 64u, (uint64_t)ldb, 4u, 3u);
        } else {
            tdm_load_2d((uint32_t)(uintptr_t)&Bsmem[buf][0],
                        B + (size_t)kb * ldb + tn, 64u, 32u, (uint64_t)ldb, 5u, 3u);
        }
    };

    // prologue: fetch first chunk and publish it
    if (wave == 0) {
        issue_chunk(0, 0);
        __builtin_amdgcn_s_wait_tensorcnt(0);
    }
    __syncthreads();

    int buf = 0;
    for (int kb = 0; kb < K; kb += 32) {
        const bool have_next = (kb + 32) < K;
        // overlap: start next chunk's DMA before computing this one
        if (wave == 0 && have_next) issue_chunk(buf ^ 1, kb + 32);

        float (*Asb)[36] = As[buf];
        float (*BsT)[36] = (float (*)[36])Bsmem[buf];
        float (*BsN)[68] = (float (*)[68])Bsmem[buf];

        // A 16x4 fp32 fragment: lane(lm,hi) VGPR v -> (M=lm, K=v+2*hi)
        // B 4x16 fragment:      lane(lm,hi) VGPR v -> (K=v+2*hi, N=lm)
        #pragma unroll
        for (int kk = 0; kk < 32; kk += 4) {
            v2f a0 = *(const v2f*)&Asb[wm + lm][kk + 2 * hi];
            v2f a1 = *(const v2f*)&Asb[wm + 16 + lm][kk + 2 * hi];
            v2f b0, b1;
            if (TRANS_B) {
                b0 = *(const v2f*)&BsT[wn + lm][kk + 2 * hi];
                b1 = *(const v2f*)&BsT[wn + 16 + lm][kk + 2 * hi];
            } else {
                b0.x = BsN[kk + 2 * hi][wn + lm];
                b0.y = BsN[kk + 2 * hi + 1][wn + lm];
                b1.x = BsN[kk + 2 * hi][wn + 16 + lm];
                b1.y = BsN[kk + 2 * hi + 1][wn + 16 + lm];
            }
            acc00 = __builtin_amdgcn_wmma_f32_16x16x4_f32(false, a0, false, b0, (short)0, acc00, false, false);
            acc01 = __builtin_amdgcn_wmma_f32_16x16x4_f32(false, a0, false, b1, (short)0, acc01, false, false);
            acc10 = __builtin_amdgcn_wmma_f32_16x16x4_f32(false, a1, false, b0, (short)0, acc10, false, false);
            acc11 = __builtin_amdgcn_wmma_f32_16x16x4_f32(false, a1, false, b1, (short)0, acc11, false, false);
        }

        // drain the in-flight DMA (wave0's TENSORcnt), then publish buffer swap
        if (wave == 0 && have_next) __builtin_amdgcn_s_wait_tensorcnt(0);
        __syncthreads();
        buf ^= 1;
    }

    // C/D layout: lane(lm,hi) VGPR v -> (M = v + 8*hi, N = lm) within 16x16 tile
    v8f accs[2][2] = {{acc00, acc01}, {acc10, acc11}};
    #pragma unroll
    for (int i = 0; i < 2; ++i) {
        #pragma unroll
        for (int j = 0; j < 2; ++j) {
            const int col  = tn + wn + 16 * j + lm;
            const int row0 = tm + wm + 16 * i + (hi << 3);
            #pragma unroll
            for (int v = 0; v < 8; ++v) {
                const int r = row0 + v;
                const size_t idx = (size_t)r * ldc + col;
                float o = alpha * accs[i][j][v];
                if (beta != 0.0f)  o += beta * C[idx];
                if (gamma != 0.0f && r == col) o += gamma;
                C[idx] = o;
            }
        }
    }
}

// ---------------------------------------------------------------------------
// Elementwise / reduction helpers
// ---------------------------------------------------------------------------
__global__ void zero_scal_kernel(float* s) {
    if (threadIdx.x == 0 && blockIdx.x == 0) { s[0] = 0.0f; }
}

__global__ __launch_bounds__(256)
void trace_sum_kernel(const float* __restrict__ A, int n, float* s) {
    __shared__ float red[256];
    int i = blockIdx.x * 256 + threadIdx.x;
    float v = (i < n) ? A[(size_t)i * n + i] : 0.0f;
    red[threadIdx.x] = v; __syncthreads();
    for (int st = 128; st > 0; st >>= 1) {
        if (threadIdx.x < st) red[threadIdx.x] += red[threadIdx.x + st];
        __syncthreads();
    }
    if (threadIdx.x == 0) atomicAdd(s, red[0]);
}

__global__ __launch_bounds__(256)
void sumsq_kernel(const float* __restrict__ A, size_t cnt, float* s) {
    __shared__ float red[256];
    float acc = 0.0f;
    for (size_t i = (size_t)blockIdx.x * 256 + threadIdx.x; i < cnt;
         i += (size_t)gridDim.x * 256)
    { float v = A[i]; acc += v * v; }
    red[threadIdx.x] = acc; __syncthreads();
    for (int st = 128; st > 0; st >>= 1) {
        if (threadIdx.x < st) red[threadIdx.x] += red[threadIdx.x + st];
        __syncthreads();
    }
    if (threadIdx.x == 0) atomicAdd(s, red[0]);
}

// mode 0: s[1] = 1/s[0]                             (inverse seed 1/trace)
// mode 1: c = sqrt(s[0]); s[1]=1/c; s[2]=sqrt(c)    (Frobenius normalization)
__global__ void prep_scalars_kernel(float* s, int mode) {
    if (threadIdx.x != 0 || blockIdx.x != 0) return;
    if (mode == 0) { s[1] = 1.0f / s[0]; }
    else { float c = sqrtf(s[0]); s[1] = 1.0f / c; s[2] = sqrtf(c); }
}

__global__ __launch_bounds__(256)
void set_identity_kernel(float* __restrict__ C, int n, const float* sp) {
    size_t tot = (size_t)n * n;
    float d = sp ? sp[0] : 1.0f;
    for (size_t i = (size_t)blockIdx.x * 256 + threadIdx.x; i < tot;
         i += (size_t)gridDim.x * 256) {
        size_t r = i / n, c = i - r * n;
        C[i] = (r == c) ? d : 0.0f;
    }
}

__global__ __launch_bounds__(256)
void scale_by_ptr_kernel(const float* __restrict__ in, float* __restrict__ out,
                         size_t cnt, const float* sp) {
    float d = sp[0];
    for (size_t i = (size_t)blockIdx.x * 256 + threadIdx.x; i < cnt;
         i += (size_t)gridDim.x * 256)
        out[i] = in[i] * d;
}

__global__ __launch_bounds__(256)
void add_inplace_kernel(float* __restrict__ a, const float* __restrict__ b, size_t cnt) {
    for (size_t i = (size_t)blockIdx.x * 256 + threadIdx.x; i < cnt;
         i += (size_t)gridDim.x * 256)
        a[i] += b[i];
}

__global__ __launch_bounds__(256)
void bias_add_kernel(float* __restrict__ h, const float* __restrict__ b,
                     size_t cnt, int d) {
    for (size_t i = (size_t)blockIdx.x * 256 + threadIdx.x; i < cnt;
         i += (size_t)gridDim.x * 256)
        h[i] += b[i % d];
}

// one block (256 threads) per row; d == 256
__global__ __launch_bounds__(256)
void rownorm_kernel(const float* __restrict__ h, float* __restrict__ hn, int d) {
    __shared__ float red[256];
    int row = blockIdx.x, t = threadIdx.x;
    float v = h[(size_t)row * d + t];
    red[t] = v * v; __syncthreads();
    for (int st = 128; st > 0; st >>= 1) {
        if (t < st) red[t] += red[t + st];
        __syncthreads();
    }
    float nrm = fmaxf(sqrtf(red[0]), 1e-8f);
    hn[(size_t)row * d + t] = v / nrm;
}

// apply  X <- X @ G(theta,0,1)^T  in place (only columns 0,1 change)
__global__ __launch_bounds__(256)
void givens_cols_kernel(float* __restrict__ X, const float* __restrict__ theta,
                        int rows, int d) {
    int r = blockIdx.x * 256 + threadIdx.x;
    if (r >= rows) return;
    float c = cosf(theta[0]), s = sinf(theta[0]);
    float t0 = X[(size_t)r * d + 0], t1 = X[(size_t)r * d + 1];
    X[(size_t)r * d + 0] = c * t0 - s * t1;
    X[(size_t)r * d + 1] = s * t0 + c * t1;
}

// per-row top-k: k iterative argmax passes over LDS-cached row, then threshold
__global__ __launch_bounds__(256)
void topk_rows_kernel(const float* __restrict__ in, float* __restrict__ out,
                      int n, int k) {
    __shared__ float buf[NROW];
    __shared__ float rmax[256];
    __shared__ int   rarg[256];
    __shared__ float thresh;
    const int row = blockIdx.x, t = threadIdx.x;
    const float* src = in + (size_t)row * n;
    for (int i = t; i < n; i += 256) buf[i] = src[i];
    __syncthreads();
    for (int it = 0; it < k; ++it) {
        float m = -INFINITY; int a = 0;
        for (int i = t; i < n; i += 256) {
            float v = buf[i];
            if (v > m) { m = v; a = i; }
        }
        rmax[t] = m; rarg[t] = a; __syncthreads();
        for (int st = 128; st > 0; st >>= 1) {
            if (t < st && rmax[t + st] > rmax[t]) {
                rmax[t] = rmax[t + st]; rarg[t] = rarg[t + st];
            }
            __syncthreads();
        }
        if (t == 0) { thresh = rmax[0]; buf[rarg[0]] = -INFINITY; }
        __syncthreads();
    }
    float th = thresh;
    float* dst = out + (size_t)row * n;
    for (int i = t; i < n; i += 256) {
        float v = src[i];
        dst[i] = (v >= th) ? v : 0.0f;
    }
}

// ---------------------------------------------------------------------------
// Host-side orchestration (all on `stream`, graph-capture safe).
// Workspace: 256 B scalars + 4 * 2 MB + 5 * 16 MB ~= 92.3 MB.
// ---------------------------------------------------------------------------
extern "C" void kernel_launch(void* const* d_in, const int* in_sizes, int n_in,
                              void* d_out, int out_size, void* d_ws, size_t ws_size,
                              hipStream_t stream) {
    const float* x     = (const float*)d_in[0];   // [2048,512]
    const float* raw   = (const float*)d_in[1];   // [2048,2048]
    const float* W     = (const float*)d_in[2];   // [256,512]
    const float* bias  = (const float*)d_in[3];   // [256]
    const float* theta = (const float*)d_in[4];   // [1]
    float* out = (float*)d_out;                   // [2048,2048]
    (void)in_sizes; (void)n_in; (void)out_size; (void)ws_size;

    char* ws = (char*)d_ws;
    float* scal = (float*)ws;                     // s[0]=reduction, s[1],s[2]=derived
    size_t off = 256;
    auto alloc = [&](size_t elems) { float* p = (float*)(ws + off); off += elems * sizeof(float); return p; };
    const size_t NN = (size_t)NROW * NROW;
    const size_t ND = (size_t)NROW * DOUT;

    float* tmpXW = alloc(ND);
    float* h     = alloc(ND);
    float* hn    = alloc(ND);
    float* Bm    = alloc(ND);
    float* B0 = alloc(NN);
    float* B1 = alloc(NN);
    float* B2 = alloc(NN);
    float* B3 = alloc(NN);
    float* B4 = alloc(NN);

    const dim3 blk(128);
    auto gemm_nn = [&](const float* A, int lda, const float* B, int ldb,
                       float* C, int ldc, int M, int Nn, int K,
                       float al, float be, float ga) {
        dim3 g(Nn / 64, M / 64);
        gemm_wmma_f32<false><<<g, blk, 0, stream>>>(A, lda, B, ldb, C, ldc, K, al, be, ga);
    };
    auto gemm_nt = [&](const float* A, int lda, const float* B, int ldb,
                       float* C, int ldc, int M, int Nn, int K,
                       float al, float be, float ga) {
        dim3 g(Nn / 64, M / 64);
        gemm_wmma_f32<true><<<g, blk, 0, stream>>>(A, lda, B, ldb, C, ldc, K, al, be, ga);
    };
    const int EW = 1024;  // elementwise grid blocks

    // 1) h = raw_adj @ (x @ W^T + b)
    gemm_nt(x, DIN, W, DIN, tmpXW, DOUT, NROW, DOUT, DIN, 1.f, 0.f, 0.f);
    bias_add_kernel<<<EW, 256, 0, stream>>>(tmpXW, bias, ND, DOUT);
    gemm_nn(raw, NROW, tmpXW, DOUT, h, DOUT, NROW, DOUT, NROW, 1.f, 0.f, 0.f);

    // 2) x_ex = inv(h h^T + 0.1 I)  via Newton-Schulz, X0 = I / trace(A)
    gemm_nt(h, DOUT, h, DOUT, B0, NROW, NROW, NROW, DOUT, 1.f, 0.f, 0.1f);
    zero_scal_kernel<<<1, 1, 0, stream>>>(scal);
    trace_sum_kernel<<<NROW / 256, 256, 0, stream>>>(B0, NROW, scal);
    prep_scalars_kernel<<<1, 1, 0, stream>>>(scal, 0);
    set_identity_kernel<<<EW, 256, 0, stream>>>(B1, NROW, scal + 1);
    {
        float* cur = B1; float* nxt = B3;
        for (int it = 0; it < INV_ITERS; ++it) {
            gemm_nn(B0, NROW, cur, NROW, B2, NROW, NROW, NROW, NROW, -1.f, 0.f, 2.f); // 2I - A X
            gemm_nn(cur, NROW, B2, NROW, nxt, NROW, NROW, NROW, NROW, 1.f, 0.f, 0.f); // X(2I-AX)
            float* t = cur; cur = nxt; nxt = t;
        }
        // INV_ITERS even -> x_ex in B1
    }
    float* x_ex = B1;

    // 3) sim_adj = top_k(hn hn^T, 20)
    rownorm_kernel<<<NROW, 256, 0, stream>>>(h, hn, DOUT);
    gemm_nt(hn, DOUT, hn, DOUT, B0, NROW, NROW, NROW, DOUT, 1.f, 0.f, 0.f);
    topk_rows_kernel<<<NROW, 256, 0, stream>>>(B0, B2, NROW, TOPK_K);  // sim in B2

    // 4) Bsum = raw^2 + sim^2  (L = 2)
    gemm_nn(raw, NROW, raw, NROW, B3, NROW, NROW, NROW, NROW, 1.f, 0.f, 0.f); // adj^2
    gemm_nn(B2, NROW, B2, NROW, B0, NROW, NROW, NROW, NROW, 1.f, 0.f, 0.f);   // sim^2
    add_inplace_kernel<<<EW, 256, 0, stream>>>(B0, B3, NN);                   // B0 = Bsum

    // 5) B = Bsum @ h @ rot^T  (Givens applied analytically to cols 0,1)
    gemm_nn(B0, NROW, h, DOUT, Bm, DOUT, NROW, DOUT, NROW, 1.f, 0.f, 0.f);
    givens_cols_kernel<<<NROW / 256, 256, 0, stream>>>(Bm, theta, NROW, DOUT);

    // 6) xb = h B^T + B h^T
    gemm_nt(h, DOUT, Bm, DOUT, B2, NROW, NROW, NROW, DOUT, 1.f, 0.f, 0.f);
    gemm_nt(Bm, DOUT, h, DOUT, B2, NROW, NROW, NROW, DOUT, 1.f, 1.f, 0.f);

    // 7) M = 0.5 (x_ex xb + xb x_ex)
    gemm_nn(x_ex, NROW, B2, NROW, B4, NROW, NROW, NROW, NROW, 0.5f, 0.f, 0.f);
    gemm_nn(B2, NROW, x_ex, NROW, B4, NROW, NROW, NROW, NROW, 0.5f, 1.f, 0.f);

    // 8) A = sqrtm(M) via coupled Newton-Schulz on M / ||M||_F
    //    (matches P diag(v^{1/L}) P^T for L=2 up to the reference's 1e-6 clip)
    zero_scal_kernel<<<1, 1, 0, stream>>>(scal);
    sumsq_kernel<<<EW, 256, 0, stream>>>(B4, NN, scal);
    prep_scalars_kernel<<<1, 1, 0, stream>>>(scal, 1);     // s1 = 1/||M||_F, s2 = ||M||_F^0.5
    scale_by_ptr_kernel<<<EW, 256, 0, stream>>>(B4, B0, NN, scal + 1); // Y0
    set_identity_kernel<<<EW, 256, 0, stream>>>(x_ex, NROW, nullptr);  // Z0 = I (reuse B1)
    {
        float* Y = B0; float* Z = x_ex; float* Yn = B4; float* Zn = B3;
        for (int it = 0; it < SQRT_ITERS; ++it) {
            gemm_nn(Z, NROW, Y, NROW, B2, NROW, NROW, NROW, NROW, -0.5f, 0.f, 1.5f); // T = 1.5I - 0.5 ZY
            gemm_nn(Y, NROW, B2, NROW, Yn, NROW, NROW, NROW, NROW, 1.f, 0.f, 0.f);   // Y <- Y T
            gemm_nn(B2, NROW, Z, NROW, Zn, NROW, NROW, NROW, NROW, 1.f, 0.f, 0.f);   // Z <- T Z
            float* t = Y; Y = Yn; Yn = t;
            t = Z; Z = Zn; Zn = t;
        }
        // SQRT_ITERS even -> Y back in B0
    }
    scale_by_ptr_kernel<<<EW, 256, 0, stream>>>(B0, B2, NN, scal + 2); // A_raw = sqrt(c) * Y

    // 9) A = top_k(A_raw, 20)
    topk_rows_kernel<<<NROW, 256, 0, stream>>>(B2, out, NROW, TOPK_K);
}